// GATv2MHA_816043786519
// MI455X (gfx1250) — compile-verified
//
#include <hip/hip_runtime.h>
#include <hip/hip_bf16.h>
#include <math.h>

// ---------------------------------------------------------------------------
// GATv2 multi-head attention, CDNA5 (gfx1250), wave32.
//   b=4, n=512, input_dim=128, heads=8, head_dim=16
// Stage 1: g_l/g_r projections  -> V_WMMA_F32_16X16X4_F32 (K-loop of 4)
// Stage 2: fused scores + softmax + (P @ g_r) per 16-row tile
//          scores are VALU (LeakyReLU breaks matmul form), P@g_r is WMMA.
// ---------------------------------------------------------------------------

#define B_SZ      4
#define N_TOK     512
#define IN_DIM    128
#define N_HEADS   8
#define HD        16
#define LEAKY     0.2f
#define NPAD      (N_TOK + 4)   // LDS row padding: stride 516 % 64 banks = 4

typedef __attribute__((ext_vector_type(2))) float v2f;
typedef __attribute__((ext_vector_type(8))) float v8f;

// ---------------------------------------------------------------------------
// Kernel 1: per (head, batch, row-tile) compute 16x16 tiles of g_l and g_r.
//   A = h rows (16 x 128), B = W (128 x 16), K-loop in steps of 4.
// One wave per block.
// ---------------------------------------------------------------------------
__global__ __launch_bounds__(32)
void gat_proj_kernel(const float* __restrict__ h,
                     const float* __restrict__ Wl,
                     const float* __restrict__ Wr,
                     float* __restrict__ gl,
                     float* __restrict__ gr) {
    const int mt   = blockIdx.x;          // row tile (n/16)
    const int b    = blockIdx.y;          // batch
    const int hh   = blockIdx.z;          // head
    const int lane = threadIdx.x;         // wave32
    const int ln15 = lane & 15;           // M (for A) / N (for B)
    const int ksh  = (lane >> 4) * 2;     // K sub-offset: lanes 16-31 hold K+2

    const float* hb = h  + ((size_t)b * N_TOK + (size_t)mt * 16) * IN_DIM;
    const float* wl = Wl + (size_t)hh * IN_DIM * HD;
    const float* wr = Wr + (size_t)hh * IN_DIM * HD;

    v8f cl = {};
    v8f cr = {};
    for (int kk = 0; kk < IN_DIM / 4; ++kk) {
        const int k0 = kk * 4 + ksh;
        v2f a, bl, br;
        // A fragment: element (M=ln15, K=k0 / k0+1)
        a.x = hb[(size_t)ln15 * IN_DIM + k0];
        a.y = hb[(size_t)ln15 * IN_DIM + k0 + 1];
        // B fragments: element (K=k0 / k0+1, N=ln15)
        bl.x = wl[(size_t)k0 * HD + ln15];
        bl.y = wl[(size_t)(k0 + 1) * HD + ln15];
        br.x = wr[(size_t)k0 * HD + ln15];
        br.y = wr[(size_t)(k0 + 1) * HD + ln15];
        cl = __builtin_amdgcn_wmma_f32_16x16x4_f32(false, a, false, bl,
                                                   (short)0, cl, false, false);
        cr = __builtin_amdgcn_wmma_f32_16x16x4_f32(false, a, false, br,
                                                   (short)0, cr, false, false);
    }

    // D layout: VGPR v holds (M = v + 8*(lane>=16), N = lane&15)
    float* glb = gl + (((size_t)hh * B_SZ + b) * N_TOK + (size_t)mt * 16) * HD;
    float* grb = gr + (((size_t)hh * B_SZ + b) * N_TOK + (size_t)mt * 16) * HD;
    const int rbase = (lane >> 4) * 8;
#pragma unroll
    for (int v = 0; v < 8; ++v) {
        const int r = rbase + v;
        glb[(size_t)r * HD + ln15] = cl[v];
        grb[(size_t)r * HD + ln15] = cr[v];
    }
}

// ---------------------------------------------------------------------------
// Kernel 2: per (head, batch, 16-row i-tile):
//   scores (VALU) -> LDS, row softmax (wave shuffles), P @ g_r via WMMA
//   with K=512 split across 4 waves (128 K each, 32 x K=4 steps), LDS reduce,
//   ReLU, store to (b, i, h*16+d).
// 128 threads = 4 waves.
// ---------------------------------------------------------------------------
__global__ __launch_bounds__(128)
void gat_attn_kernel(const float* __restrict__ gl,
                     const float* __restrict__ gr,
                     const float* __restrict__ Wak,
                     const unsigned char* __restrict__ mask,
                     float* __restrict__ out) {
    const int it   = blockIdx.x;          // i tile (n/16)
    const int b    = blockIdx.y;
    const int hh   = blockIdx.z;
    const int tid  = threadIdx.x;
    const int lane = tid & 31;
    const int wave = tid >> 5;

    __shared__ float s_scores[16 * NPAD];     // scores then probs
    __shared__ float s_gl[16 * HD];
    __shared__ float s_wak[HD];
    __shared__ float s_part[4][256];          // per-wave partial C tiles

    const float* glb = gl + (((size_t)hh * B_SZ + b) * N_TOK + (size_t)it * 16) * HD;
    const float* grb = gr + ((size_t)hh * B_SZ + b) * N_TOK * HD;
    const unsigned char* mrow =
        mask + ((size_t)b * N_TOK + (size_t)it * 16) * N_TOK;

    if (tid < HD) s_wak[tid] = Wak[(size_t)hh * HD + tid];
    for (int idx = tid; idx < 16 * HD; idx += 128) s_gl[idx] = glb[idx];
    __syncthreads();

    // ---- scores: each thread owns j values, reuses gr[j] across all 16 i ----
    for (int j = tid; j < N_TOK; j += 128) {
        float grv[HD];
#pragma unroll
        for (int d = 0; d < HD; ++d) grv[d] = grb[(size_t)j * HD + d];
#pragma unroll
        for (int i = 0; i < 16; ++i) {
            float s = 0.f;
#pragma unroll
            for (int d = 0; d < HD; ++d) {
                const float x = s_gl[i * HD + d] + grv[d];
                s += s_wak[d] * ((x > 0.f) ? x : LEAKY * x);
            }
            if (mrow[(size_t)i * N_TOK + j]) s = -INFINITY;
            s_scores[i * NPAD + j] = s;
        }
    }
    __syncthreads();

    // ---- softmax: wave w handles rows w, w+4, w+8, w+12 ----
    for (int i = wave; i < 16; i += 4) {
        float m = -INFINITY;
        for (int j = lane; j < N_TOK; j += 32)
            m = fmaxf(m, s_scores[i * NPAD + j]);
#pragma unroll
        for (int o = 16; o > 0; o >>= 1) m = fmaxf(m, __shfl_xor(m, o, 32));
        float sum = 0.f;
        for (int j = lane; j < N_TOK; j += 32) {
            const float e = __expf(s_scores[i * NPAD + j] - m);
            s_scores[i * NPAD + j] = e;
            sum += e;
        }
#pragma unroll
        for (int o = 16; o > 0; o >>= 1) sum += __shfl_xor(sum, o, 32);
        const float inv = 1.f / sum;
        for (int j = lane; j < N_TOK; j += 32) s_scores[i * NPAD + j] *= inv;
    }
    __syncthreads();

    // ---- out tile = P(16x512) @ g_r(512x16); wave w covers K=[w*128,w*128+128)
    v8f c = {};
    const int rowA = lane & 15;
    const int ksh  = (lane >> 4) * 2;
    for (int kk = 0; kk < 32; ++kk) {
        const int j0 = wave * 128 + kk * 4 + ksh;
        v2f a, bb;
        a.x  = s_scores[rowA * NPAD + j0];
        a.y  = s_scores[rowA * NPAD + j0 + 1];
        bb.x = grb[(size_t)j0 * HD + rowA];
        bb.y = grb[(size_t)(j0 + 1) * HD + rowA];
        c = __builtin_amdgcn_wmma_f32_16x16x4_f32(false, a, false, bb,
                                                  (short)0, c, false, false);
    }
#pragma unroll
    for (int v = 0; v < 8; ++v) s_part[wave][v * 32 + lane] = c[v];
    __syncthreads();

    if (wave == 0) {
#pragma unroll
        for (int v = 0; v < 8; ++v) {
            float r = s_part[0][v * 32 + lane] + s_part[1][v * 32 + lane] +
                      s_part[2][v * 32 + lane] + s_part[3][v * 32 + lane];
            r = fmaxf(r, 0.f);  // ReLU
            const int rr = v + 8 * (lane >> 4);   // local i
            const int dd = lane & 15;             // head-dim
            out[((size_t)b * N_TOK + (size_t)it * 16 + rr) * (N_HEADS * HD) +
                (size_t)hh * HD + dd] = r;
        }
    }
}

// ---------------------------------------------------------------------------
extern "C" void kernel_launch(void* const* d_in, const int* in_sizes, int n_in,
                              void* d_out, int out_size, void* d_ws, size_t ws_size,
                              hipStream_t stream) {
    (void)in_sizes; (void)n_in; (void)out_size; (void)ws_size;
    const float*         h    = (const float*)d_in[0];
    const unsigned char* mask = (const unsigned char*)d_in[1];
    const float*         Wl   = (const float*)d_in[2];
    const float*         Wr   = (const float*)d_in[3];
    const float*         Wak  = (const float*)d_in[4];
    float* out = (float*)d_out;

    float* gl = (float*)d_ws;                                  // (H,B,N,HD)
    float* gr = gl + (size_t)N_HEADS * B_SZ * N_TOK * HD;      // (H,B,N,HD)

    dim3 grid(N_TOK / 16, B_SZ, N_HEADS);
    gat_proj_kernel<<<grid, 32, 0, stream>>>(h, Wl, Wr, gl, gr);
    gat_attn_kernel<<<grid, 128, 0, stream>>>(gl, gr, Wak, mask, out);
}